// GATConv_26216480375291
// MI455X (gfx1250) — compile-verified
//
#include <hip/hip_runtime.h>

typedef __attribute__((ext_vector_type(2))) float v2f;
typedef __attribute__((ext_vector_type(8))) float v8f;

#define IN_FEATS 256
#define HEADS 8
#define DHEAD 32
#define HD 256          // HEADS*DHEAD
#define ALPHA 0.2f

// ---------------------------------------------------------------------------
// Kernel 1: ft = feat @ W.T  via V_WMMA_F32_16X16X4_F32 (exact fp32)
// One block = 128 threads = 4 waves; block owns a 16-row M tile x all 256 cols.
// Wave w computes 4 consecutive 16-col N tiles (cols 64w .. 64w+63).
// ---------------------------------------------------------------------------
__global__ __launch_bounds__(128) void gat_gemm_kernel(
    const float* __restrict__ feat, const float* __restrict__ W,
    float* __restrict__ ft, int n_nodes) {
  const int lane  = threadIdx.x & 31;
  const int wave  = threadIdx.x >> 5;     // 0..3
  const int row0  = blockIdx.x << 4;      // 16 rows / block
  const int m     = lane & 15;
  const int khalf = lane >> 4;            // 0 or 1

  int arow = row0 + m;
  if (arow >= n_nodes) arow = n_nodes - 1;   // clamp loads; stores guarded

  const int c0 = wave << 2;               // first of 4 col-tiles for this wave
  const float* Abase = feat + (size_t)arow * IN_FEATS + 2 * khalf;
  const float* B0 = W + (size_t)(((c0 + 0) << 4) + m) * IN_FEATS + 2 * khalf;
  const float* B1 = W + (size_t)(((c0 + 1) << 4) + m) * IN_FEATS + 2 * khalf;
  const float* B2 = W + (size_t)(((c0 + 2) << 4) + m) * IN_FEATS + 2 * khalf;
  const float* B3 = W + (size_t)(((c0 + 3) << 4) + m) * IN_FEATS + 2 * khalf;

  v8f acc0 = {}; v8f acc1 = {}; v8f acc2 = {}; v8f acc3 = {};

  for (int k = 0; k < IN_FEATS; k += 4) {
    v2f a  = *(const v2f*)(Abase + k);
    v2f b0 = *(const v2f*)(B0 + k);
    v2f b1 = *(const v2f*)(B1 + k);
    v2f b2 = *(const v2f*)(B2 + k);
    v2f b3 = *(const v2f*)(B3 + k);
    acc0 = __builtin_amdgcn_wmma_f32_16x16x4_f32(false, a, false, b0, (short)0, acc0, false, false);
    acc1 = __builtin_amdgcn_wmma_f32_16x16x4_f32(false, a, false, b1, (short)0, acc1, false, false);
    acc2 = __builtin_amdgcn_wmma_f32_16x16x4_f32(false, a, false, b2, (short)0, acc2, false, false);
    acc3 = __builtin_amdgcn_wmma_f32_16x16x4_f32(false, a, false, b3, (short)0, acc3, false, false);
  }

  const int col0 = ((c0 + 0) << 4) + m;
  const int col1 = ((c0 + 1) << 4) + m;
  const int col2 = ((c0 + 2) << 4) + m;
  const int col3 = ((c0 + 3) << 4) + m;
  const int rbase = row0 + 8 * khalf;
#pragma unroll
  for (int g = 0; g < 8; ++g) {
    int r = rbase + g;
    if (r < n_nodes) {
      size_t off = (size_t)r * HD;
      ft[off + col0] = acc0[g];
      ft[off + col1] = acc1[g];
      ft[off + col2] = acc2[g];
      ft[off + col3] = acc3[g];
    }
  }
}

// ---------------------------------------------------------------------------
// Kernel 2: el/er per (node, head): 32-length dot with attn vectors
// gid = n*8 + h;  ft offset = gid*32
// ---------------------------------------------------------------------------
__global__ void gat_elr_kernel(const float* __restrict__ ft,
                               const float* __restrict__ attn_l,
                               const float* __restrict__ attn_r,
                               float* __restrict__ el, float* __restrict__ er,
                               int total) {
  int gid = blockIdx.x * blockDim.x + threadIdx.x;
  if (gid >= total) return;
  int h = gid & 7;
  const float* f  = ft + (size_t)gid * DHEAD;
  const float* al = attn_l + h * DHEAD;
  const float* ar = attn_r + h * DHEAD;
  float sl = 0.f, sr = 0.f;
#pragma unroll
  for (int d = 0; d < DHEAD; ++d) {
    float x = f[d];
    sl += x * al[d];
    sr += x * ar[d];
  }
  el[gid] = sl;
  er[gid] = sr;
}

// ---------------------------------------------------------------------------
// CSR construction: histogram, single-block scan, scatter
// ---------------------------------------------------------------------------
__global__ void zero_i32_kernel(int* __restrict__ p, int n) {
  int i = blockIdx.x * blockDim.x + threadIdx.x;
  if (i < n) p[i] = 0;
}

__global__ void count_kernel(const int* __restrict__ dst, int* __restrict__ counts, int E) {
  int i = blockIdx.x * blockDim.x + threadIdx.x;
  if (i < E) atomicAdd(&counts[dst[i]], 1);
}

__global__ __launch_bounds__(1024) void scan_kernel(const int* __restrict__ counts,
                                                    int* __restrict__ offsets,
                                                    int* __restrict__ cursor, int n) {
  __shared__ int buf[1024];
  __shared__ int carry_s;
  const int t = threadIdx.x;
  if (t == 0) carry_s = 0;
  __syncthreads();
  for (int base = 0; base < n; base += 1024) {
    int i = base + t;
    int v = (i < n) ? counts[i] : 0;
    buf[t] = v;
    __syncthreads();
    for (int off = 1; off < 1024; off <<= 1) {
      int x = (t >= off) ? buf[t - off] : 0;
      __syncthreads();
      buf[t] += x;
      __syncthreads();
    }
    int excl  = buf[t] - v;
    int carry = carry_s;
    if (i < n) { offsets[i] = carry + excl; cursor[i] = carry + excl; }
    int total = buf[1023];
    __syncthreads();
    if (t == 0) carry_s = carry + total;
    __syncthreads();
  }
  if (t == 0) offsets[n] = carry_s;
}

__global__ void scatter_kernel(const int* __restrict__ src, const int* __restrict__ dst,
                               int* __restrict__ cursor, int* __restrict__ src_sorted, int E) {
  int i = blockIdx.x * blockDim.x + threadIdx.x;
  if (i < E) {
    int p = atomicAdd(&cursor[dst[i]], 1);
    src_sorted[p] = src[i];
  }
}

// ---------------------------------------------------------------------------
// Kernel 7: per-dst-node edge softmax + weighted aggregation.
// One 256-thread block per node. ft gathers are coalesced 1KB reads that
// should hit L2 (ft = 102MB < 192MB). Output written exactly once (no atomics).
// ---------------------------------------------------------------------------
__global__ __launch_bounds__(256) void gat_aggregate_kernel(
    const float* __restrict__ ft, const float* __restrict__ el,
    const float* __restrict__ er, const int* __restrict__ src_sorted,
    const int* __restrict__ offsets, float* __restrict__ out, int n_nodes) {
  const int n = blockIdx.x;
  const int t = threadIdx.x;
  __shared__ float s_er[HEADS];
  __shared__ float s_hmax[HEADS];
  __shared__ float s_hsum[HEADS];
  __shared__ float s_red[64];
  __shared__ float s_a[256];    // [32 edges][8 heads]
  __shared__ int   s_src[32];

  const int start = offsets[n];
  const int end   = offsets[n + 1];

  if (t < HEADS) s_er[t] = er[(size_t)n * HEADS + t];
  __syncthreads();

  // ---- pass 1: per-head max over incoming edges (8 slots per head) ----
  if (t < 64) {
    int h = t & 7, slot = t >> 3;
    float er_h = s_er[h];
    float mx = -3.402823466e38f;
    for (int i = start + slot; i < end; i += 8) {
      float v = el[(size_t)src_sorted[i] * HEADS + h] + er_h;
      v = v > 0.f ? v : ALPHA * v;
      mx = fmaxf(mx, v);
    }
    s_red[t] = mx;
  }
  __syncthreads();
  if (t < 32) s_red[t] = fmaxf(s_red[t], s_red[t + 32]);
  __syncthreads();
  if (t < 16) s_red[t] = fmaxf(s_red[t], s_red[t + 16]);
  __syncthreads();
  if (t < 8)  s_hmax[t] = fmaxf(s_red[t], s_red[t + 8]);
  __syncthreads();

  // ---- pass 2: per-head sum of exp(v - max) ----
  if (t < 64) {
    int h = t & 7, slot = t >> 3;
    float er_h = s_er[h], mx = s_hmax[h];
    float s = 0.f;
    for (int i = start + slot; i < end; i += 8) {
      float v = el[(size_t)src_sorted[i] * HEADS + h] + er_h;
      v = v > 0.f ? v : ALPHA * v;
      s += __expf(v - mx);
    }
    s_red[t] = s;
  }
  __syncthreads();
  if (t < 32) s_red[t] += s_red[t + 32];
  __syncthreads();
  if (t < 16) s_red[t] += s_red[t + 16];
  __syncthreads();
  if (t < 8)  s_hsum[t] = fmaxf(s_red[t] + s_red[t + 8], 1e-20f);
  __syncthreads();

  // ---- pass 3: weighted aggregation; thread t owns output element (h,d) ----
  const int h = t >> 5;
  float acc = 0.f;
  for (int chunk = start; chunk < end; chunk += 32) {
    int cnt = min(32, end - chunk);
    if (t < 32) s_src[t] = (t < cnt) ? src_sorted[chunk + t] : 0;
    __syncthreads();
    {
      int ei = t >> 3, hh = t & 7;
      float aval = 0.f;
      if (ei < cnt) {
        float v = el[(size_t)s_src[ei] * HEADS + hh] + s_er[hh];
        v = v > 0.f ? v : ALPHA * v;
        aval = __expf(v - s_hmax[hh]) / s_hsum[hh];
      }
      s_a[(ei << 3) + hh] = aval;
    }
    __syncthreads();
    for (int ei = 0; ei < cnt; ++ei)
      acc = fmaf(s_a[(ei << 3) + h], ft[(size_t)s_src[ei] * HD + t], acc);
    __syncthreads();
  }
  out[(size_t)n * HD + t] = acc;
}

// ---------------------------------------------------------------------------
extern "C" void kernel_launch(void* const* d_in, const int* in_sizes, int n_in,
                              void* d_out, int out_size, void* d_ws, size_t ws_size,
                              hipStream_t stream) {
  const float* feat   = (const float*)d_in[0];
  const float* W      = (const float*)d_in[1];
  const float* attn_l = (const float*)d_in[2];
  const float* attn_r = (const float*)d_in[3];
  const int*   src    = (const int*)d_in[4];
  const int*   dst    = (const int*)d_in[5];

  const int N = in_sizes[0] / IN_FEATS;
  const int E = in_sizes[4];

  char* ws = (char*)d_ws;
  size_t off = 0;
  auto take = [&](size_t bytes) -> char* {
    char* p = ws + off;
    off = (off + bytes + 255) & ~(size_t)255;
    return p;
  };
  float* ft         = (float*)take((size_t)N * HD * sizeof(float));
  float* el         = (float*)take((size_t)N * HEADS * sizeof(float));
  float* er         = (float*)take((size_t)N * HEADS * sizeof(float));
  int*   counts     = (int*)take((size_t)N * sizeof(int));
  int*   offsets    = (int*)take((size_t)(N + 1) * sizeof(int));
  int*   cursor     = (int*)take((size_t)N * sizeof(int));
  int*   src_sorted = (int*)take((size_t)E * sizeof(int));
  (void)ws_size; (void)n_in; (void)out_size;

  gat_gemm_kernel<<<(N + 15) / 16, 128, 0, stream>>>(feat, W, ft, N);
  gat_elr_kernel<<<(N * HEADS + 255) / 256, 256, 0, stream>>>(ft, attn_l, attn_r, el, er, N * HEADS);
  zero_i32_kernel<<<(N + 255) / 256, 256, 0, stream>>>(counts, N);
  count_kernel<<<(E + 255) / 256, 256, 0, stream>>>(dst, counts, E);
  scan_kernel<<<1, 1024, 0, stream>>>(counts, offsets, cursor, N);
  scatter_kernel<<<(E + 255) / 256, 256, 0, stream>>>(src, dst, cursor, src_sorted, E);
  gat_aggregate_kernel<<<N, 256, 0, stream>>>(ft, el, er, src_sorted, offsets, (float*)d_out, N);
}